// RoIBBox_45715631899301
// MI455X (gfx1250) — compile-verified
//
#include <hip/hip_runtime.h>
#include <hip/hip_bf16.h>

#define FM 64
#define AC 9
#define NBOX (FM * FM * AC)   // 36864
#define BATCH 16
#define PRE_NMS 1000
#define POST_NMS 300
#define CAND 1024
#define IOU_TH 0.7f

typedef __attribute__((ext_vector_type(2))) float v2f;
typedef __attribute__((ext_vector_type(8))) float v8f;
typedef __attribute__((ext_vector_type(4))) int   v4i;

#define AS1 __attribute__((address_space(1)))
#define AS3 __attribute__((address_space(3)))

// ---------------------------------------------------------------------------
// Kernel 1: box decode. Per-lane compute of [cy,cx,h,w], then the constant
// 4x4 corner transform is done as D = A(16x4,=T^T padded) x B(4x16,=chw^T)
// via V_WMMA_F32_16X16X4_F32. D[m][n] (m<4) = coord m of box n, so lanes
// 0..15 hold y1,x1,y2,x2 of one box in d[0..3] -> one 16B store per lane.
// Exact: T entries are {0, +-0.5, 1}.
// ---------------------------------------------------------------------------
__global__ __launch_bounds__(256) void decode_kernel(
    const float* __restrict__ deltas,   // (B*NBOX, 4)
    const float* __restrict__ anchors,  // (NBOX, 4)
    float* __restrict__ boxesWs)        // (B*NBOX, 4)
{
    const int g = blockIdx.x * 256 + threadIdx.x;   // one box per lane
    const int n = g % NBOX;

    const float4 dl = *(const float4*)(deltas + (size_t)g * 4);
    const float4 an = *(const float4*)(anchors + (size_t)n * 4);

    const float ah  = an.z - an.x;
    const float aw  = an.w - an.y;
    const float acy = an.x + 0.5f * ah;
    const float acx = an.y + 0.5f * aw;

    const float cy = (dl.x * 0.1f) * ah + acy;
    const float cx = (dl.y * 0.1f) * aw + acx;
    const float h  = __expf(dl.z * 0.2f) * ah;
    const float w  = __expf(dl.w * 0.2f) * aw;

    const int lane = threadIdx.x & 31;
    const int m    = lane & 15;
    const bool hi  = lane >= 16;

    // A = T^T padded to 16x4. Lanes 0-15: VGPR0=K0, VGPR1=K1; lanes 16-31: K2,K3.
    float a0, a1;
    if (!hi) {  // K=0 (cy coeff), K=1 (cx coeff)
        a0 = (m == 0 || m == 2) ? 1.0f : 0.0f;
        a1 = (m == 1 || m == 3) ? 1.0f : 0.0f;
    } else {    // K=2 (h coeff), K=3 (w coeff)
        a0 = (m == 0) ? -0.5f : ((m == 2) ? 0.5f : 0.0f);
        a1 = (m == 1) ? -0.5f : ((m == 3) ? 0.5f : 0.0f);
    }
    v2f A; A[0] = a0; A[1] = a1;

    const int waveBase = g & ~31;   // first box of this wave

#pragma unroll
    for (int g2 = 0; g2 < 2; ++g2) {
        const int src = (g2 << 4) | m;
        const float s_cy = __shfl(cy, src, 32);
        const float s_cx = __shfl(cx, src, 32);
        const float s_h  = __shfl(h,  src, 32);
        const float s_w  = __shfl(w,  src, 32);

        // B = chw^T (4x16): lanes 0-15 rows K0/K1 (cy,cx), lanes 16-31 rows K2/K3 (h,w)
        v2f Bv; Bv[0] = hi ? s_h : s_cy; Bv[1] = hi ? s_w : s_cx;

        v8f C = {0.f, 0.f, 0.f, 0.f, 0.f, 0.f, 0.f, 0.f};
        v8f D = __builtin_amdgcn_wmma_f32_16x16x4_f32(
            false, A, false, Bv, (short)0, C, false, false);

        if (!hi) {
            const int box = waveBase + (g2 << 4) + m;
            float4 out; out.x = D[0]; out.y = D[1]; out.z = D[2]; out.w = D[3];
            *(float4*)(boxesWs + (size_t)box * 4) = out;
        }
    }
}

// ---------------------------------------------------------------------------
// Kernel 2: per-batch top-1000. Two-pass 12-bit radix select on float bit
// patterns (scores are uniform [0,1) => positive => uint order == float
// order), candidate compaction (<=1024), LDS bitonic sort descending of
// packed (score_bits<<32)|idx keys, then gather the top-1000 boxes.
// ---------------------------------------------------------------------------
__global__ __launch_bounds__(1024) void topk_kernel(
    const float* __restrict__ labels,   // (B, NBOX)
    const float* __restrict__ boxesWs,  // (B*NBOX, 4)
    float* __restrict__ preBoxes)       // (B, PRE_NMS, 4)
{
    __shared__ unsigned hist[4096];
    __shared__ unsigned long long keys[CAND];
    __shared__ unsigned s_bin1, s_above1, s_t24, s_cnt;

    const int batch = blockIdx.x;
    const int tid   = threadIdx.x;
    const float* sc = labels + (size_t)batch * NBOX;

    // ---- pass 1: histogram of bits[31:20] ----
    for (int i = tid; i < 4096; i += 1024) hist[i] = 0;
    __syncthreads();
    for (int i = tid; i < NBOX; i += 1024) {
        unsigned u = __float_as_uint(sc[i]);
        atomicAdd(&hist[u >> 20], 1u);
    }
    __syncthreads();
    if (tid == 0) {
        unsigned c = 0;
        for (int b = 4095; b >= 0; --b) {
            unsigned nc = c + hist[b];
            if (nc >= PRE_NMS) { s_bin1 = (unsigned)b; s_above1 = c; break; }
            c = nc;
        }
        s_cnt = 0;
    }
    __syncthreads();
    const unsigned bin1 = s_bin1, above1 = s_above1;

    // ---- pass 2: refine on bits[19:8] within boundary bin ----
    for (int i = tid; i < 4096; i += 1024) hist[i] = 0;
    __syncthreads();
    for (int i = tid; i < NBOX; i += 1024) {
        unsigned u = __float_as_uint(sc[i]);
        if ((u >> 20) == bin1) atomicAdd(&hist[(u >> 8) & 0xFFFu], 1u);
    }
    __syncthreads();
    if (tid == 0) {
        unsigned c = above1;
        for (int b = 4095; b >= 0; --b) {
            unsigned nc = c + hist[b];
            if (nc >= PRE_NMS) { s_t24 = (bin1 << 12) | (unsigned)b; break; }
            c = nc;
        }
    }
    __syncthreads();
    const unsigned t24 = s_t24;

    // ---- compact candidates ----
    for (int i = tid; i < NBOX; i += 1024) {
        unsigned u = __float_as_uint(sc[i]);
        if ((u >> 8) >= t24) {
            unsigned slot = atomicAdd(&s_cnt, 1u);
            if (slot < CAND)
                keys[slot] = ((unsigned long long)u << 32) | (unsigned)i;
        }
    }
    __syncthreads();
    unsigned cnt = s_cnt; if (cnt > CAND) cnt = CAND;
    if ((unsigned)tid >= cnt) keys[tid] = 0ull;   // pad with lowest key
    __syncthreads();

    // ---- bitonic sort descending, 1024 keys ----
    for (int k = 2; k <= CAND; k <<= 1) {
        for (int j = k >> 1; j > 0; j >>= 1) {
            int ixj = tid ^ j;
            if (ixj > tid) {
                unsigned long long a = keys[tid], b = keys[ixj];
                bool up = ((tid & k) == 0);
                if (up ? (a < b) : (a > b)) { keys[tid] = b; keys[ixj] = a; }
            }
            __syncthreads();
        }
    }

    // ---- gather top-1000 boxes ----
    if (tid < PRE_NMS) {
        unsigned idx = (unsigned)(keys[tid] & 0xFFFFFFFFull);
        float4 bx = *(const float4*)(boxesWs + ((size_t)batch * NBOX + idx) * 4);
        *(float4*)(preBoxes + ((size_t)batch * PRE_NMS + tid) * 4) = bx;
    }
}

// ---------------------------------------------------------------------------
// Kernel 3: suppression bitmask. mask[batch][i][w] bit b = (j = w*32+b) > i
// && j < 1000 && IoU(i,j) > 0.7. Boxes staged in LDS once per block via the
// gfx1250 async global->LDS path (ASYNCcnt) when available.
// Block = 8 rows x 32 words; grid = BATCH * 125.
// ---------------------------------------------------------------------------
__global__ __launch_bounds__(256) void nms_mask_kernel(
    const float* __restrict__ preBoxes,
    unsigned* __restrict__ maskWs)
{
    __shared__ float4 sb[PRE_NMS];
    const int batch = blockIdx.x / 125;
    const int grp   = blockIdx.x % 125;
    const float4* pb = (const float4*)(preBoxes + (size_t)batch * PRE_NMS * 4);

#if __has_builtin(__builtin_amdgcn_global_load_async_to_lds_b128)
    for (int i = threadIdx.x; i < PRE_NMS; i += 256) {
        __builtin_amdgcn_global_load_async_to_lds_b128(
            (AS1 v4i*)(AS1 void*)(float4*)(pb + i),
            (AS3 v4i*)(AS3 void*)&sb[i], 0, 0);
    }
  #if __has_builtin(__builtin_amdgcn_s_wait_asynccnt)
    __builtin_amdgcn_s_wait_asynccnt(0);
  #else
    asm volatile("s_wait_asynccnt 0x0" ::: "memory");
  #endif
#else
    for (int i = threadIdx.x; i < PRE_NMS; i += 256) sb[i] = pb[i];
#endif
    __syncthreads();

    const int i = grp * 8 + (threadIdx.x >> 5);
    const int w = threadIdx.x & 31;
    const float4 bi = sb[i];
    const float areai = (bi.z - bi.x) * (bi.w - bi.y);

    unsigned m = 0;
    const int j0 = w << 5;
#pragma unroll 4
    for (int b = 0; b < 32; ++b) {
        const int j = j0 + b;
        if (j < PRE_NMS && j > i) {
            const float4 bj = sb[j];
            const float iy1 = fmaxf(bi.x, bj.x);
            const float ix1 = fmaxf(bi.y, bj.y);
            const float iy2 = fminf(bi.z, bj.z);
            const float ix2 = fminf(bi.w, bj.w);
            const float inter = fmaxf(iy2 - iy1, 0.f) * fmaxf(ix2 - ix1, 0.f);
            const float areaj = (bj.z - bj.x) * (bj.w - bj.y);
            const float uni = fmaxf(areai + areaj - inter, 1e-9f);
            if (inter > IOU_TH * uni) m |= (1u << b);
        }
    }
    maskWs[((size_t)batch * PRE_NMS + i) * 32 + w] = m;
}

// ---------------------------------------------------------------------------
// Kernel 4: greedy scan + output. One wave per batch: lane w owns suppression
// word w; 1000 sequential steps with wave-broadcast of the decision bit.
// Then lane 0 compacts kept boxes (clipped to [0,1]) into the first <=300
// slots and zero-pads the rest (d_out fully rewritten every call).
// ---------------------------------------------------------------------------
__global__ __launch_bounds__(32) void nms_scan_kernel(
    const unsigned* __restrict__ maskWs,
    const float* __restrict__ preBoxes,
    float* __restrict__ out)
{
    __shared__ unsigned keepw[32];
    const int batch = blockIdx.x;
    const int lane  = threadIdx.x;
    const unsigned* M = maskWs + (size_t)batch * PRE_NMS * 32;

    unsigned removed = 0;
    for (int i = 0; i < PRE_NMS; ++i) {
        if (i + 1 < PRE_NMS)
            __builtin_prefetch(&M[(size_t)(i + 1) * 32 + lane], 0, 3);
        const unsigned rw = __shfl(removed, i >> 5, 32);
        if (!((rw >> (i & 31)) & 1u))
            removed |= M[(size_t)i * 32 + lane];
    }
    const unsigned valid = (lane < 31) ? 0xFFFFFFFFu : 0x000000FFu; // boxes 992..999
    keepw[lane] = (~removed) & valid;
    __syncthreads();

    if (lane == 0) {
        float* o = out + (size_t)batch * POST_NMS * 4;
        int cnt = 0;
        for (int i = 0; i < PRE_NMS && cnt < POST_NMS; ++i) {
            if ((keepw[i >> 5] >> (i & 31)) & 1u) {
                const float* b = preBoxes + ((size_t)batch * PRE_NMS + i) * 4;
                o[cnt * 4 + 0] = fminf(fmaxf(b[0], 0.f), 1.f);
                o[cnt * 4 + 1] = fminf(fmaxf(b[1], 0.f), 1.f);
                o[cnt * 4 + 2] = fminf(fmaxf(b[2], 0.f), 1.f);
                o[cnt * 4 + 3] = fminf(fmaxf(b[3], 0.f), 1.f);
                ++cnt;
            }
        }
        for (; cnt < POST_NMS; ++cnt) {
            o[cnt * 4 + 0] = 0.f; o[cnt * 4 + 1] = 0.f;
            o[cnt * 4 + 2] = 0.f; o[cnt * 4 + 3] = 0.f;
        }
    }
}

// ---------------------------------------------------------------------------
extern "C" void kernel_launch(void* const* d_in, const int* in_sizes, int n_in,
                              void* d_out, int out_size, void* d_ws, size_t ws_size,
                              hipStream_t stream) {
    const float* deltas  = (const float*)d_in[0];   // (16,64,64,36) f32
    const float* labels  = (const float*)d_in[1];   // (16,64,64,9)  f32
    const float* anchors = (const float*)d_in[2];   // (36864,4)     f32
    float* out = (float*)d_out;                     // (16,300,4)    f32

    // workspace layout
    float* boxesWs  = (float*)d_ws;                                   // 16*36864*4 f32
    float* preBoxes = boxesWs + (size_t)BATCH * NBOX * 4;             // 16*1000*4  f32
    unsigned* maskWs = (unsigned*)(preBoxes + (size_t)BATCH * PRE_NMS * 4); // 16*1000*32 u32

    const int totalBoxes = BATCH * NBOX;            // 589824, multiple of 256
    decode_kernel<<<totalBoxes / 256, 256, 0, stream>>>(deltas, anchors, boxesWs);
    topk_kernel<<<BATCH, 1024, 0, stream>>>(labels, boxesWs, preBoxes);
    nms_mask_kernel<<<BATCH * 125, 256, 0, stream>>>(preBoxes, maskWs);
    nms_scan_kernel<<<BATCH, 32, 0, stream>>>(maskWs, preBoxes, out);
}